// StickBreakingSegmentation_34557306864319
// MI455X (gfx1250) — compile-verified
//
#include <hip/hip_runtime.h>
#include <math.h>

#define BB     32
#define CIN    64
#define COUT   8
#define IMG    128
#define HW     (IMG * IMG)      // 16384
#define KSTEPS 8

typedef __attribute__((ext_vector_type(2))) float v2f;
typedef __attribute__((ext_vector_type(8))) float v8f;

// ---------------------------------------------------------------------------
// Kernel 1: feat[b][o][p] = gate * (sum_c W[o][c]*x[b][c][p] + bias[o]) + uv
// via V_WMMA_F32_16X16X4_F32.  One wave per 16-pixel tile:
//   A (16x4)  = x tile      (M = pixel-in-tile, K = 4 input channels)
//   B (4x16)  = W^T padded  (K = input channel, N = output channel, rows 8..15 = 0)
//   D (16x16) = per-pixel per-channel partial sums, accumulated over 16 k-steps.
// D layout: VGPR v -> pixels pbase+v (lanes 0-15) / pbase+v+8 (lanes 16-31),
// N = lane&15.  So lane (o = lane&15 < 8) holds 8 consecutive pixels of
// channel o across acc[0..7] -> two contiguous float4 stores.
// Also zero-initializes scopes[:,0,:] with lanes whose o == 8.
// ---------------------------------------------------------------------------
__global__ __launch_bounds__(256) void semiconv_wmma_kernel(
    const float* __restrict__ x, const float* __restrict__ conv_w,
    const float* __restrict__ conv_b, const float* __restrict__ gate_p,
    float* __restrict__ feat, float* __restrict__ scopes) {
  __shared__ float wpad[16 * CIN];  // W zero-padded to 16 output rows
  for (int i = threadIdx.x; i < 16 * CIN; i += 256) {
    int row = i >> 6, col = i & 63;
    wpad[i] = (row < COUT) ? conv_w[row * CIN + col] : 0.0f;
  }
  __syncthreads();

  const int lane  = threadIdx.x & 31;
  const int wave  = (blockIdx.x * 256 + threadIdx.x) >> 5;
  const int b     = wave >> 10;              // 1024 tiles per batch image
  const int pbase = (wave & 1023) << 4;      // 16 pixels, never crosses a row

  const int m     = lane & 15;               // pixel-in-tile (A) / out-chan (B,D)
  const int khalf = lane >> 4;               // K half select (ISA A/B layout)

  const float* xb = x + ((size_t)b * CIN) * HW + pbase;

  v8f acc = {};
#pragma unroll
  for (int k0 = 0; k0 < CIN; k0 += 4) {
    const int cA = k0 + khalf * 2;           // this lane's K pair
    v2f a;
    a.x = xb[(size_t)cA * HW + m];           // A: K=cA,   M=m
    a.y = xb[(size_t)(cA + 1) * HW + m];     // A: K=cA+1, M=m
    v2f bm;
    {                                         // B: K=cA/cA+1, N=m (from LDS, padded)
      float2 wv = *(const float2*)&wpad[m * CIN + cA];
      bm.x = wv.x;
      bm.y = wv.y;
    }
    acc = __builtin_amdgcn_wmma_f32_16x16x4_f32(
        /*neg_a=*/false, a, /*neg_b=*/false, bm,
        /*c_mod=*/(short)0, acc, /*reuse_a=*/false, /*reuse_b=*/false);
  }

  const float gate  = *gate_p;
  const int o       = lane & 15;
  const int pixoff  = (lane >> 4) << 3;      // 0 or 8
  const int p0      = pbase + pixoff;

  if (o < COUT) {
    const float bias = conv_b[o];
    const int h  = p0 >> 7;                  // constant within tile
    const int w0 = p0 & 127;
    float vals[8];
#pragma unroll
    for (int v = 0; v < 8; ++v) {
      float val = gate * (acc[v] + bias);
      if (o == 6) val += -1.0f + (float)h * (2.0f / 127.0f);        // g1 (row)
      if (o == 7) val += -1.0f + (float)(w0 + v) * (2.0f / 127.0f); // g2 (col)
      vals[v] = val;
    }
    float* dst = feat + ((size_t)b * COUT + o) * HW + p0;
    *(float4*)(dst)     = make_float4(vals[0], vals[1], vals[2], vals[3]);
    *(float4*)(dst + 4) = make_float4(vals[4], vals[5], vals[6], vals[7]);
  } else if (o == COUT) {                    // lanes 8 & 24: scope_0 = 0
    float* s0 = scopes + (size_t)b * KSTEPS * HW + p0;
    *(float4*)(s0)     = make_float4(0.f, 0.f, 0.f, 0.f);
    *(float4*)(s0 + 4) = make_float4(0.f, 0.f, 0.f, 0.f);
  }
}

// ---------------------------------------------------------------------------
// Kernel 2: per-batch argmax of rand * exp(scope_k) (first-occurrence ties),
// then gather the 8-channel seed feature vector into ws.
// ---------------------------------------------------------------------------
__global__ __launch_bounds__(256) void argmax_seed_kernel(
    const float* __restrict__ rand_pixel, const float* __restrict__ scopes,
    const float* __restrict__ feat, float* __restrict__ seed, int k) {
  __shared__ float sv[256];
  __shared__ int   si[256];
  const int b = blockIdx.x;
  const int t = threadIdx.x;
  const float* r = rand_pixel + (size_t)b * HW;
  const float* s = scopes + ((size_t)b * KSTEPS + k) * HW;

  float best = -1.0f;  // all candidate values are >= 0
  int   bi   = 0;
  for (int p = t; p < HW; p += 256) {        // ascending p -> strict > keeps first max
    float v = r[p] * expf(s[p]);
    if (v > best) { best = v; bi = p; }
  }
  sv[t] = best; si[t] = bi;
  __syncthreads();
  for (int off = 128; off > 0; off >>= 1) {
    if (t < off) {
      float ov = sv[t + off]; int oi = si[t + off];
      if (ov > sv[t] || (ov == sv[t] && oi < si[t])) { sv[t] = ov; si[t] = oi; }
    }
    __syncthreads();
  }
  if (t < COUT) {
    const int idx = si[0];
    seed[b * COUT + t] = feat[((size_t)b * COUT + t) * HW + idx];
  }
}

// ---------------------------------------------------------------------------
// Kernel 3: one stick-breaking step, float4 over pixels.
//   alpha = clip(exp(-||feat-seed||^2 / sigma), .01, .99)
//   masks[b][k]   = scope + log(alpha)
//   scopes[b][k+1]= scope + log1p(-alpha)   (on k==6 also masks[b][7] = it)
// ---------------------------------------------------------------------------
__global__ __launch_bounds__(256) void stick_step_kernel(
    const float* __restrict__ feat, const float* __restrict__ seed,
    const float* __restrict__ log_sigma_p,
    float* __restrict__ masks, float* __restrict__ scopes, int k) {
  const int gid   = blockIdx.x * 256 + threadIdx.x;
  const int per_b = HW / 4;                  // 4096 float4's per batch
  const int b     = gid / per_b;             // uniform per block (16 blocks/batch)
  const int p     = (gid - b * per_b) * 4;

  const float inv_sigma = expf(-log_sigma_p[0]);  // 1/sigma

  float sd[COUT];
#pragma unroll
  for (int o = 0; o < COUT; ++o) sd[o] = seed[b * COUT + o];

  const float* fb = feat + (size_t)b * COUT * HW + p;
  float d0 = 0.f, d1 = 0.f, d2 = 0.f, d3 = 0.f;
#pragma unroll
  for (int o = 0; o < COUT; ++o) {
    float4 f = *(const float4*)(fb + (size_t)o * HW);
    float t0 = f.x - sd[o]; d0 += t0 * t0;
    float t1 = f.y - sd[o]; d1 += t1 * t1;
    float t2 = f.z - sd[o]; d2 += t2 * t2;
    float t3 = f.w - sd[o]; d3 += t3 * t3;
  }

  const size_t bk = ((size_t)b * KSTEPS + k) * HW + p;
  float4 s = *(const float4*)(scopes + bk);

  float4 lm, sn;
  {
    float a;
    a = fminf(fmaxf(expf(-d0 * inv_sigma), 0.01f), 0.99f);
    lm.x = s.x + logf(a); sn.x = s.x + log1pf(-a);
    a = fminf(fmaxf(expf(-d1 * inv_sigma), 0.01f), 0.99f);
    lm.y = s.y + logf(a); sn.y = s.y + log1pf(-a);
    a = fminf(fmaxf(expf(-d2 * inv_sigma), 0.01f), 0.99f);
    lm.z = s.z + logf(a); sn.z = s.z + log1pf(-a);
    a = fminf(fmaxf(expf(-d3 * inv_sigma), 0.01f), 0.99f);
    lm.w = s.w + logf(a); sn.w = s.w + log1pf(-a);
  }

  *(float4*)(masks + bk)       = lm;
  *(float4*)(scopes + bk + HW) = sn;         // scopes[b][k+1]
  if (k == KSTEPS - 2) {                     // last step: masks[b][7] = final scope
    *(float4*)(masks + ((size_t)b * KSTEPS + (KSTEPS - 1)) * HW + p) = sn;
  }
}

extern "C" void kernel_launch(void* const* d_in, const int* in_sizes, int n_in,
                              void* d_out, int out_size, void* d_ws, size_t ws_size,
                              hipStream_t stream) {
  const float* x          = (const float*)d_in[0];  // [32,64,128,128]
  const float* rand_pixel = (const float*)d_in[1];  // [32,1,128,128]
  const float* conv_w     = (const float*)d_in[2];  // [8,64]
  const float* conv_b     = (const float*)d_in[3];  // [8]
  const float* gate       = (const float*)d_in[4];  // scalar
  const float* log_sigma  = (const float*)d_in[5];  // scalar

  float* masks  = (float*)d_out;                          // [B,K,H,W]
  float* scopes = masks + (size_t)BB * KSTEPS * HW;       // [B,K,H,W]

  float* feat = (float*)d_ws;                             // [B,8,HW]  (16 MiB)
  float* seed = feat + (size_t)BB * COUT * HW;            // [B,8]

  // feat + scope_0 init: 32768 tiles, 8 waves/block -> 4096 blocks
  semiconv_wmma_kernel<<<4096, 256, 0, stream>>>(x, conv_w, conv_b, gate,
                                                 feat, scopes);
  for (int k = 0; k < KSTEPS - 1; ++k) {
    argmax_seed_kernel<<<BB, 256, 0, stream>>>(rand_pixel, scopes, feat, seed, k);
    stick_step_kernel<<<(BB * HW / 4) / 256, 256, 0, stream>>>(
        feat, seed, log_sigma, masks, scopes, k);
  }
}